// InteractionBlock_34445637714614
// MI455X (gfx1250) — compile-verified
//
#include <hip/hip_runtime.h>

// ---------------------------------------------------------------------------
// Equivariant GNN interaction block, fused for MI455X (gfx1250).
//
// Phases:
//   1) up_s / up_v : fp32 WMMA GEMMs (V_WMMA_F32_16X16X4_F32) -> up[N,256]
//   2) edge_kernel : gather up[senders], fused edge math, scatter with
//                    hardware f32 atomics into acc_s[N,128], acc_v[N,384]
//   3) dn_s / dn_v : fp32 WMMA GEMMs on accumulators (x 1/16) -> out[N,256]
//
// The 51MB up-table and 102MB accumulators are L2-resident (192MB L2), so
// the random gather/scatter phase runs at L2 bandwidth; only ~400MB touches
// HBM (23.3 TB/s -> ~20us floor).
// ---------------------------------------------------------------------------

typedef __attribute__((ext_vector_type(2))) float v2f;
typedef __attribute__((ext_vector_type(4))) float v4f;
typedef __attribute__((ext_vector_type(8))) float v8f;

constexpr int   NN        = 50000;   // nodes
constexpr int   NE        = 800000;  // edges
constexpr float INV_SQRT3 = 0.57735026918962576451f;
constexpr float INV_AVG   = 1.0f / 16.0f;   // 1 / AVG_NEIGH

// 4-byte-aligned 3-float aggregate (HIP's float3 may carry wider alignment).
struct __attribute__((packed, aligned(4))) f3 { float x, y, z; };

static __device__ __forceinline__ v8f wmma_f32_k4(v8f c, v2f a, v2f b) {
  // D = A(16x4,f32) * B(4x16,f32) + C(16x16,f32)
  return __builtin_amdgcn_wmma_f32_16x16x4_f32(
      /*neg_a=*/false, a, /*neg_b=*/false, b,
      /*c_mod=*/(short)0, c, /*reuse_a=*/false, /*reuse_b=*/false);
}

// ---------------------------------------------------------------------------
// Up-projection, scalar channels:  up[n, d] = sum_c nf[n, c] * W[c, d]
// Block = 128 threads = 4 waves; wave w owns output column tile d = 16w..16w+15.
// blockIdx.x owns node rows 16*bx .. 16*bx+15   (50000 % 16 == 0).
// ---------------------------------------------------------------------------
__global__ __launch_bounds__(128) void up_s_kernel(
    const float* __restrict__ nf, const float* __restrict__ W,
    float* __restrict__ up) {
  const int lane = threadIdx.x & 31;
  const int wave = threadIdx.x >> 5;
  const int m    = lane & 15;   // row within tile (A) / col within tile (B,D)
  const int hi   = lane >> 4;   // half-wave selects K pair (A,B) / M+8 (D)
  const int r0   = blockIdx.x * 16;
  const int n0   = wave * 16;
  const float* __restrict__ arow = nf + (size_t)(r0 + m) * 256;  // s in cols 0..63
  v8f c = {};
#pragma unroll
  for (int k0 = 0; k0 < 64; k0 += 4) {
    const int ka = k0 + 2 * hi;            // lanes 0-15: K=k0,k0+1 ; 16-31: K=k0+2,k0+3
    v2f a, b;
    a.x = arow[ka];
    a.y = arow[ka + 1];
    b.x = W[ka * 64 + n0 + m];             // B row ka, col n0+m
    b.y = W[(ka + 1) * 64 + n0 + m];
    c = wmma_f32_k4(c, a, b);
  }
#pragma unroll
  for (int r = 0; r < 8; ++r) {            // D: VGPR r -> row r + 8*hi, col m
    const int M = r + 8 * hi;
    up[(size_t)(r0 + M) * 256 + n0 + m] = c[r];
  }
}

// ---------------------------------------------------------------------------
// Up-projection, vector channels: up[n, 64 + d*3 + i] = sum_c v[n,c,i]*W[c,d]
// GEMM row space is r = n*3 + i (150000 rows, % 16 == 0); A element (r,k) sits
// at nf[n*256 + 64 + 3k + i]  (stride-3 loads, absorbed by L1/L2).
// ---------------------------------------------------------------------------
__global__ __launch_bounds__(128) void up_v_kernel(
    const float* __restrict__ nf, const float* __restrict__ W,
    float* __restrict__ up) {
  const int lane = threadIdx.x & 31;
  const int wave = threadIdx.x >> 5;
  const int m    = lane & 15;
  const int hi   = lane >> 4;
  const int r0   = blockIdx.x * 16;
  const int n0   = wave * 16;
  const int row  = r0 + m;
  const int n    = row / 3;
  const int i    = row - 3 * n;
  const float* __restrict__ arow = nf + (size_t)n * 256 + 64 + i;  // k at arow[3k]
  v8f c = {};
#pragma unroll
  for (int k0 = 0; k0 < 64; k0 += 4) {
    const int ka = k0 + 2 * hi;
    v2f a, b;
    a.x = arow[3 * ka];
    a.y = arow[3 * (ka + 1)];
    b.x = W[ka * 64 + n0 + m];
    b.y = W[(ka + 1) * 64 + n0 + m];
    c = wmma_f32_k4(c, a, b);
  }
#pragma unroll
  for (int r = 0; r < 8; ++r) {
    const int rr = r0 + r + 8 * hi;
    const int nn = rr / 3;
    const int ii = rr - 3 * nn;
    up[(size_t)nn * 256 + 64 + (n0 + m) * 3 + ii] = c[r];
  }
}

// ---------------------------------------------------------------------------
// Edge phase: one thread per (edge, channel). 64 consecutive lanes share an
// edge -> coalesced gather of the sender row; edge_attrs as one b128 load;
// 8 hardware f32 atomics / thread into L2-resident accumulators.
// ---------------------------------------------------------------------------
__global__ __launch_bounds__(256) void edge_kernel(
    const float* __restrict__ up, const float* __restrict__ ea,
    const int* __restrict__ snd, const int* __restrict__ rcv,
    float* __restrict__ acc_s, float* __restrict__ acc_v) {
  const int tid = blockIdx.x * 256 + threadIdx.x;
  const int e   = tid >> 6;
  const int ch  = tid & 63;
  const int s   = snd[e];
  const int r   = rcv[e];
  const v4f y   = ((const v4f*)ea)[e];     // single global_load_b128 (16B aligned)
  const float* __restrict__ un = up + (size_t)s * 256;
  const float se = un[ch];
  const f3    ve = ((const f3*)(un + 64))[ch];   // 12B, candidate for b96
  float* as = acc_s + (size_t)r * 128;
  float* av = acc_v + (size_t)r * 384;
  unsafeAtomicAdd(&as[ch],      se * y.x);                                       // m_s0
  unsafeAtomicAdd(&as[64 + ch], (ve.x * y.y + ve.y * y.z + ve.z * y.w) * INV_SQRT3); // m_s1
  unsafeAtomicAdd(&av[ch * 3 + 0], ve.x * y.x);                                  // m_v0
  unsafeAtomicAdd(&av[ch * 3 + 1], ve.y * y.x);
  unsafeAtomicAdd(&av[ch * 3 + 2], ve.z * y.x);
  unsafeAtomicAdd(&av[(64 + ch) * 3 + 0], se * y.y);                             // m_v1
  unsafeAtomicAdd(&av[(64 + ch) * 3 + 1], se * y.z);
  unsafeAtomicAdd(&av[(64 + ch) * 3 + 2], se * y.w);
}

// ---------------------------------------------------------------------------
// Down-projection, scalar: out[n, d] = (1/16) * sum_c acc_s[n,c] * W[c,d], K=128
// ---------------------------------------------------------------------------
__global__ __launch_bounds__(128) void dn_s_kernel(
    const float* __restrict__ acc_s, const float* __restrict__ W,
    float* __restrict__ out) {
  const int lane = threadIdx.x & 31;
  const int wave = threadIdx.x >> 5;
  const int m    = lane & 15;
  const int hi   = lane >> 4;
  const int r0   = blockIdx.x * 16;
  const int n0   = wave * 16;
  const float* __restrict__ arow = acc_s + (size_t)(r0 + m) * 128;
  v8f c = {};
#pragma unroll
  for (int k0 = 0; k0 < 128; k0 += 4) {
    const int ka = k0 + 2 * hi;
    v2f a, b;
    a.x = arow[ka];
    a.y = arow[ka + 1];
    b.x = W[ka * 64 + n0 + m];
    b.y = W[(ka + 1) * 64 + n0 + m];
    c = wmma_f32_k4(c, a, b);
  }
#pragma unroll
  for (int r = 0; r < 8; ++r) {
    const int M = r + 8 * hi;
    out[(size_t)(r0 + M) * 256 + n0 + m] = c[r] * INV_AVG;
  }
}

// ---------------------------------------------------------------------------
// Down-projection, vector: out[n, 64 + d*3 + i] = (1/16)*sum_c acc_v[n,c,i]W[c,d]
// ---------------------------------------------------------------------------
__global__ __launch_bounds__(128) void dn_v_kernel(
    const float* __restrict__ acc_v, const float* __restrict__ W,
    float* __restrict__ out) {
  const int lane = threadIdx.x & 31;
  const int wave = threadIdx.x >> 5;
  const int m    = lane & 15;
  const int hi   = lane >> 4;
  const int r0   = blockIdx.x * 16;
  const int n0   = wave * 16;
  const int row  = r0 + m;
  const int n    = row / 3;
  const int i    = row - 3 * n;
  const float* __restrict__ arow = acc_v + (size_t)n * 384 + i;  // k at arow[3k]
  v8f c = {};
#pragma unroll
  for (int k0 = 0; k0 < 128; k0 += 4) {
    const int ka = k0 + 2 * hi;
    v2f a, b;
    a.x = arow[3 * ka];
    a.y = arow[3 * (ka + 1)];
    b.x = W[ka * 64 + n0 + m];
    b.y = W[(ka + 1) * 64 + n0 + m];
    c = wmma_f32_k4(c, a, b);
  }
#pragma unroll
  for (int r = 0; r < 8; ++r) {
    const int rr = r0 + r + 8 * hi;
    const int nn = rr / 3;
    const int ii = rr - 3 * nn;
    out[(size_t)nn * 256 + 64 + (n0 + m) * 3 + ii] = c[r] * INV_AVG;
  }
}

// ---------------------------------------------------------------------------
extern "C" void kernel_launch(void* const* d_in, const int* in_sizes, int n_in,
                              void* d_out, int out_size, void* d_ws, size_t ws_size,
                              hipStream_t stream) {
  const float* nf  = (const float*)d_in[0];  // node_feats (N,256)
  const float* ea  = (const float*)d_in[1];  // edge_attrs (E,4)
  const float* Wus = (const float*)d_in[2];  // W_up_s (64,64)
  const float* Wuv = (const float*)d_in[3];  // W_up_v (64,64)
  const float* Wds = (const float*)d_in[4];  // W_dn_s (128,64)
  const float* Wdv = (const float*)d_in[5];  // W_dn_v (128,64)
  const int*   snd = (const int*)d_in[6];    // senders (E)
  const int*   rcv = (const int*)d_in[7];    // receivers (E)
  float* out = (float*)d_out;                // (N,256)

  const size_t acc_s_bytes = (size_t)NN * 128 * sizeof(float);  // 25.6 MB
  const size_t acc_v_bytes = (size_t)NN * 384 * sizeof(float);  // 76.8 MB
  const size_t up_bytes    = (size_t)NN * 256 * sizeof(float);  // 51.2 MB

  char*  ws    = (char*)d_ws;
  float* acc_s = (float*)ws;
  float* acc_v = (float*)(ws + acc_s_bytes);
  // Up-table: in workspace if it fits, else reuse d_out as scratch (it is
  // dead by the time dn_* kernels overwrite d_out; stream order guarantees
  // edge_kernel finished reading it first).
  float* up = (ws_size >= acc_s_bytes + acc_v_bytes + up_bytes)
                  ? (float*)(ws + acc_s_bytes + acc_v_bytes)
                  : out;

  // Zero the segment-sum accumulators every call (deterministic).
  hipMemsetAsync(acc_s, 0, acc_s_bytes + acc_v_bytes, stream);

  up_s_kernel<<<NN / 16,        128, 0, stream>>>(nf, Wus, up);
  up_v_kernel<<<(3 * NN) / 16,  128, 0, stream>>>(nf, Wuv, up);
  edge_kernel<<<(NE * 64) / 256, 256, 0, stream>>>(up, ea, snd, rcv, acc_s, acc_v);
  dn_s_kernel<<<NN / 16,        128, 0, stream>>>(acc_s, Wds, out);
  dn_v_kernel<<<(3 * NN) / 16,  128, 0, stream>>>(acc_v, Wdv, out);
}